// TripletLoss_90091234001212
// MI455X (gfx1250) — compile-verified
//
#include <hip/hip_runtime.h>
#include <math.h>

// Triplet margin loss, MI455X (gfx1250, wave32).
// Pass 1: one wave per row, b128 non-temporal streaming loads, FMA accumulate,
//         cross-lane reduction done on the matrix pipe (V_WMMA_F32_16X16X4_F32
//         with a selector B-matrix reduces dp and dn simultaneously).
// Pass 2: single block, deterministic tree reduce of 16384 row losses -> mean,
//         final 32-lane collapse again via WMMA ones-reduction.

typedef float v2f __attribute__((ext_vector_type(2)));
typedef float v4f __attribute__((ext_vector_type(4)));
typedef float v8f __attribute__((ext_vector_type(8)));

#define TL_BATCH 16384
#define TL_FEAT  2048
#define TL_MARGIN 1.0f

__global__ __launch_bounds__(256) void triplet_rowdist_kernel(
    const float* __restrict__ anchor,
    const float* __restrict__ positive,
    const float* __restrict__ negative,
    float* __restrict__ row_loss)
{
    const int lane = threadIdx.x & 31;
    const int wave = threadIdx.x >> 5;
    const int row  = blockIdx.x * 8 + wave;   // 2048 blocks * 8 waves = 16384 rows

    const v4f* __restrict__ A = (const v4f*)(anchor   + (size_t)row * TL_FEAT);
    const v4f* __restrict__ P = (const v4f*)(positive + (size_t)row * TL_FEAT);
    const v4f* __restrict__ N = (const v4f*)(negative + (size_t)row * TL_FEAT);

    float dp = 0.0f, dn = 0.0f;

    // FEAT/4 = 512 float4 per row; 32 lanes -> 16 iterations. Streaming data
    // (403MB > 192MB L2, zero reuse) -> non-temporal hint on every load.
#pragma unroll 4
    for (int k = 0; k < (TL_FEAT / 4) / 32; ++k) {
        const int i = k * 32 + lane;
        v4f av = __builtin_nontemporal_load(&A[i]);
        v4f pv = __builtin_nontemporal_load(&P[i]);
        v4f nv = __builtin_nontemporal_load(&N[i]);
        float d;
        d = av[0] - pv[0]; dp = fmaf(d, d, dp);
        d = av[1] - pv[1]; dp = fmaf(d, d, dp);
        d = av[2] - pv[2]; dp = fmaf(d, d, dp);
        d = av[3] - pv[3]; dp = fmaf(d, d, dp);
        d = av[0] - nv[0]; dn = fmaf(d, d, dn);
        d = av[1] - nv[1]; dn = fmaf(d, d, dn);
        d = av[2] - nv[2]; dn = fmaf(d, d, dn);
        d = av[3] - nv[3]; dn = fmaf(d, d, dn);
    }

    // Matrix-pipe cross-lane reduction.
    // A (16x4 f32): lane m<16 holds A[m][0]=dp, A[m][1]=dn; lane m+16 holds
    //               A[m][2]=dp, A[m][3]=dn  ==> a2 = {dp, dn} in every lane.
    // B (4x16): col0 = (1,0,1,0) sums dp halves, col1 = (0,1,0,1) sums dn.
    v2f a2; a2[0] = dp; a2[1] = dn;
    v2f b2;
    b2[0] = (lane == 0 || lane == 16) ? 1.0f : 0.0f;
    b2[1] = (lane == 1 || lane == 17) ? 1.0f : 0.0f;
    v8f c = {};
    c = __builtin_amdgcn_wmma_f32_16x16x4_f32(false, a2, false, b2,
                                              (short)0, c, false, false);
    // D[m][0] = dp[m]+dp[m+16] lives in lanes 0 (M=0..7) / 16 (M=8..15);
    // D[m][1] = dn[...] in lanes 1 / 17. Sum the 8 C VGPRs per lane, fold.
    float s = c[0] + c[1] + c[2] + c[3] + c[4] + c[5] + c[6] + c[7];
    s += __shfl_xor(s, 16, 32);           // lane0: dp_total, lane1: dn_total
    float o = __shfl_xor(s, 1, 32);       // lane0 now also sees dn_total

    if (lane == 0) {
        float loss = sqrtf(s) - sqrtf(o) + TL_MARGIN;
        row_loss[row] = loss > 0.0f ? loss : 0.0f;
    }
}

__global__ __launch_bounds__(256) void triplet_mean_kernel(
    const float* __restrict__ row_loss,
    float* __restrict__ out)
{
    __shared__ float sm[256];
    float s = 0.0f;
    for (int i = threadIdx.x; i < TL_BATCH; i += 256)   // fixed order: deterministic
        s += row_loss[i];
    sm[threadIdx.x] = s;
    __syncthreads();

    if (threadIdx.x < 32) {                  // wave-uniform branch, EXEC all-ones inside
        float t = 0.0f;
#pragma unroll
        for (int j = 0; j < 8; ++j) t += sm[threadIdx.x + 32 * j];

        // WMMA ones-reduction: A[m][0]=t[m] (lanes 0-15), A[m][2]=t[m+16],
        // A[m][1]=A[m][3]=0; B = all ones  =>  D[m][n] = t[m] + t[m+16].
        v2f a2; a2[0] = t;    a2[1] = 0.0f;
        v2f b2; b2[0] = 1.0f; b2[1] = 1.0f;
        v8f c = {};
        c = __builtin_amdgcn_wmma_f32_16x16x4_f32(false, a2, false, b2,
                                                  (short)0, c, false, false);
        float r = c[0] + c[1] + c[2] + c[3] + c[4] + c[5] + c[6] + c[7];
        r += __shfl_xor(r, 16, 32);          // lane0: grand total

        if (threadIdx.x == 0)
            out[0] = r * (1.0f / (float)TL_BATCH);
    }
}

extern "C" void kernel_launch(void* const* d_in, const int* in_sizes, int n_in,
                              void* d_out, int out_size, void* d_ws, size_t ws_size,
                              hipStream_t stream) {
    (void)in_sizes; (void)n_in; (void)out_size; (void)ws_size;

    const float* anchor   = (const float*)d_in[0];
    const float* positive = (const float*)d_in[1];
    const float* negative = (const float*)d_in[2];
    float* out      = (float*)d_out;
    float* row_loss = (float*)d_ws;          // 16384 floats = 64 KB of scratch

    dim3 grid1(TL_BATCH / 8);                // 8 waves (rows) per 256-thread block
    dim3 block1(256);
    triplet_rowdist_kernel<<<grid1, block1, 0, stream>>>(anchor, positive, negative, row_loss);

    triplet_mean_kernel<<<1, 256, 0, stream>>>(row_loss, out);
}